// DiffuseRouter_86835648790917
// MI455X (gfx1250) — compile-verified
//
#include <hip/hip_runtime.h>

// final[i] = (1/3) * sum of 28 expert slices at i, f32 end-to-end.
// Reduction runs on the WMMA pipe: D = A(16x4, ones) x B(4x16, data) + C,
// so D[m,n] = sum_k B[k,n]. Each wave owns 4 consecutive 16-wide tiles so
// address math is amortized: one address pair per slice group, tile loads
// use +64/+128/+192B immediate offsets. Pure HBM-bound (~304 MB, ~13us floor).

typedef float v2f __attribute__((ext_vector_type(2)));
typedef float v8f __attribute__((ext_vector_type(8)));

#define N_ELEMS 2621440   // 8 * 256 * 1280  (one expert slice / the output)
#define TILES_PER_WAVE 4  // 64 outputs per wave

__device__ __forceinline__ v8f wmma_sum4(v2f ones, v2f b, v8f c) {
    return __builtin_amdgcn_wmma_f32_16x16x4_f32(false, ones, false, b,
                                                 (short)0, c, false, false);
}

// One group of 4 expert slices starting at slice s of tensor p:
// 8 NT loads (2 address pairs, imm offsets) + 4 chained WMMAs.
__device__ __forceinline__ void accum_group(const float* __restrict__ p, int s,
                                            int khalf, size_t col0, v2f ones,
                                            v8f& a0, v8f& a1, v8f& a2, v8f& a3) {
    const size_t N = (size_t)N_ELEMS;
    const float* px = p + (size_t)(s + 2 * khalf) * N + col0;   // slices s / s+2
    const float* py = px + N;                                   // slices s+1 / s+3
    v2f b0, b1, b2, b3;
    b0.x = __builtin_nontemporal_load(px +  0); b0.y = __builtin_nontemporal_load(py +  0);
    b1.x = __builtin_nontemporal_load(px + 16); b1.y = __builtin_nontemporal_load(py + 16);
    b2.x = __builtin_nontemporal_load(px + 32); b2.y = __builtin_nontemporal_load(py + 32);
    b3.x = __builtin_nontemporal_load(px + 48); b3.y = __builtin_nontemporal_load(py + 48);
    a0 = wmma_sum4(ones, b0, a0);
    a1 = wmma_sum4(ones, b1, a1);
    a2 = wmma_sum4(ones, b2, a2);
    a3 = wmma_sum4(ones, b3, a3);
}

__global__ __launch_bounds__(256) void diffuse_router_wmma_sum(
    const float* __restrict__ e0,   // [ 4][N]
    const float* __restrict__ e1,   // [ 8][N]
    const float* __restrict__ e2,   // [16][N]
    float* __restrict__ out)        // [N]
{
    const int lane  = (int)(threadIdx.x & 31u);
    const int wave  = (int)blockIdx.x * 8 + (int)(threadIdx.x >> 5);

    const int n     = lane & 15;    // output column within a 16-wide tile
    const int khalf = lane >> 4;    // 0: loads slices s,s+1   1: slices s+2,s+3

    const size_t col0 = (size_t)wave * (16 * TILES_PER_WAVE) + (size_t)n;

    v8f a0 = {0.f,0.f,0.f,0.f,0.f,0.f,0.f,0.f};
    v8f a1 = a0, a2 = a0, a3 = a0;
    const v2f ones = {1.0f, 1.0f};

    // 28 slices = 7 groups of 4: level0 (4 experts), level1 (8), level2 (16)
    accum_group(e0, 0,  khalf, col0, ones, a0, a1, a2, a3);
    accum_group(e1, 0,  khalf, col0, ones, a0, a1, a2, a3);
    accum_group(e1, 4,  khalf, col0, ones, a0, a1, a2, a3);
    accum_group(e2, 0,  khalf, col0, ones, a0, a1, a2, a3);
    accum_group(e2, 4,  khalf, col0, ones, a0, a1, a2, a3);
    accum_group(e2, 8,  khalf, col0, ones, a0, a1, a2, a3);
    accum_group(e2, 12, khalf, col0, ones, a0, a1, a2, a3);

    // D replicated across M: lanes 0..15 hold (M=0, N=lane) in element 0.
    if (lane < 16) {
        const float w = 1.0f / 3.0f;
        float* o = out + col0;
        __builtin_nontemporal_store(a0[0] * w, o +  0);
        __builtin_nontemporal_store(a1[0] * w, o + 16);
        __builtin_nontemporal_store(a2[0] * w, o + 32);
        __builtin_nontemporal_store(a3[0] * w, o + 48);
    }
}

extern "C" void kernel_launch(void* const* d_in, const int* in_sizes, int n_in,
                              void* d_out, int out_size, void* d_ws, size_t ws_size,
                              hipStream_t stream) {
    // setup_inputs() order:
    //   0: time_emb (unused — routing is uniform since enable_time=False)
    //   1: expert_emb_0 [4,8,256,1280] f32
    //   2: expert_emb_1 [8,8,256,1280] f32
    //   3: expert_emb_2 [16,8,256,1280] f32
    //   4: time_step (unused)   5: total_steps (unused)
    (void)in_sizes; (void)n_in; (void)out_size; (void)d_ws; (void)ws_size;

    const float* e0 = (const float*)d_in[1];
    const float* e1 = (const float*)d_in[2];
    const float* e2 = (const float*)d_in[3];
    float* out = (float*)d_out;

    // 2,621,440 outputs / 64 per wave = 40,960 waves; 8 waves per block
    // -> 5,120 blocks. Exact coverage: no bounds branches, EXEC all-ones
    // through every WMMA.
    const int tiles         = N_ELEMS / 16;                    // 163840
    const int wavesTotal    = tiles / TILES_PER_WAVE;          // 40960
    dim3 block(256);
    dim3 grid(wavesTotal / 8);                                 // 5120

    diffuse_router_wmma_sum<<<grid, block, 0, stream>>>(e0, e1, e2, out);
}